// NUFNO2d_34325378629688
// MI455X (gfx1250) — compile-verified
//
#include <hip/hip_runtime.h>
#include <math.h>

// ---------------- problem constants ----------------
#define NB    8      // batch
#define NSD   16     // subdomains
#define NCH   64
#define SDC   4      // NCH/NSD
#define NMODE 512    // 32*16 truncated modes
#define NPTS  16384
#define NPAIR 992    // 32*31 (x,y) basis modes after Hermitian extension

typedef float v2f __attribute__((ext_vector_type(2)));
typedef float v8f __attribute__((ext_vector_type(8)));

__device__ __forceinline__ v8f wmma4(v2f a, v2f b, v8f c) {
  // D(16x16 f32) = A(16x4) * B(4x16) + C   -- v_wmma_f32_16x16x4_f32
  return __builtin_amdgcn_wmma_f32_16x16x4_f32(false, a, false, b, (short)0, c, false, false);
}

__device__ __forceinline__ float gelu_f(float x) {
  return 0.5f * x * (1.0f + erff(x * 0.70710678118654752f));
}

#define TWO_PI 6.283185307179586f

// ======================================================================
// Encoder / main-branch embedding kernels (elementwise, coalesced)
// ======================================================================
__global__ __launch_bounds__(256) void k_embed_sd(
    const float* __restrict__ u_sd, const float* __restrict__ sd_info,
    const float* __restrict__ wf, const float* __restrict__ bf,
    float* __restrict__ out) {
  int idx = blockIdx.x * 256 + threadIdx.x;        // (bs, c, hw)
  if (idx >= NB * NSD * SDC * 4096) return;
  int hw = idx & 4095;
  int t  = idx >> 12;
  int c  = t & 3;
  int bs = t >> 2;                                  // b*16+s
  int h = hw >> 6, w = hw & 63;
  const float* si = sd_info + (size_t)bs * 4;
  float u  = u_sd[(size_t)bs * 4096 + hw];
  float gx = si[0] + (h * (1.0f / 63.0f)) * si[2];
  float gy = si[1] + (w * (1.0f / 63.0f)) * si[3];
  float v = u * wf[0 * 4 + c] + gx * wf[1 * 4 + c] + gy * wf[2 * 4 + c] + bf[c];
  out[((size_t)bs * 4 + c) * 4096 + hw] = v;
}

__global__ __launch_bounds__(256) void k_embed_g(
    const float* __restrict__ u_g, const float* __restrict__ wf,
    const float* __restrict__ bf, float* __restrict__ out) {
  int idx = blockIdx.x * 256 + threadIdx.x;        // (b, c, hw)
  if (idx >= NB * NCH * 4096) return;
  int hw = idx & 4095;
  int t  = idx >> 12;
  int c  = t & 63;
  int b  = t >> 6;
  int h = hw >> 6, w = hw & 63;
  float u = u_g[(size_t)b * 4096 + hw];
  float v = u * wf[0 * 64 + c] + (h * (1.0f / 63.0f)) * wf[1 * 64 + c]
            + (w * (1.0f / 63.0f)) * wf[2 * 64 + c] + bf[c];
  out[((size_t)b * 64 + c) * 4096 + hw] = v;
}

// ======================================================================
// Truncated forward rDFT2: 64x64 real -> 32x16 complex modes
// (rows 0..15 = ky 0..15, rows 16..31 = ky 48..63).  One WG per image.
// ======================================================================
__global__ __launch_bounds__(256) void k_rfft(
    const float* __restrict__ in, float* __restrict__ F, int nimg) {
  __shared__ float  sImg[4096];
  __shared__ float2 sA[1024];   // [h][kx]
  __shared__ float2 sTw[64];
  int n = blockIdx.x;
  if (n >= nimg) return;
  int tid = threadIdx.x;
  const float* ip = in + (size_t)n * 4096;
  for (int t = tid; t < 4096; t += 256) sImg[t] = ip[t];
  if (tid < 64) {
    float ang = -TWO_PI * (float)tid * (1.0f / 64.0f);
    float s, c; __sincosf(ang, &s, &c);
    sTw[tid] = make_float2(c, s);
  }
  __syncthreads();
  // stage 1: DFT along w (only kx 0..15)
  for (int t = tid; t < 1024; t += 256) {
    int h = t >> 4, kx = t & 15;
    float ar = 0.f, ai = 0.f;
    for (int w = 0; w < 64; ++w) {
      float2 tw = sTw[(w * kx) & 63];
      float u = sImg[h * 64 + w];
      ar += u * tw.x; ai += u * tw.y;
    }
    sA[t] = make_float2(ar, ai);
  }
  __syncthreads();
  // stage 2: DFT along h at ky in {0..15, 48..63}
  for (int t = tid; t < NMODE; t += 256) {
    int xm = t >> 4, kx = t & 15;
    int ky = (xm < 16) ? xm : (xm + 32);
    float fr = 0.f, fi = 0.f;
    for (int h = 0; h < 64; ++h) {
      float2 a = sA[h * 16 + kx];
      float2 tw = sTw[(h * ky) & 63];
      fr += a.x * tw.x - a.y * tw.y;
      fi += a.x * tw.y + a.y * tw.x;
    }
    size_t o = ((size_t)n * NMODE + t) * 2;
    F[o] = fr; F[o + 1] = fi;
  }
}

// ======================================================================
// Spectral mode mixing: out[bb,o,m] = sum_i in[bb,i,m] * W{1,2}[i,o,m]
// One WG per mode m (512 WGs).  Generic over (nb, ci, co).
// ======================================================================
__global__ __launch_bounds__(256) void k_specmix(
    const float* __restrict__ Fin, const float* __restrict__ W1,
    const float* __restrict__ W2, float* __restrict__ Fout,
    int nb, int ci, int co) {
  __shared__ float2 sIn[512];    // nb*ci <= 512
  __shared__ float2 sW[4096];    // ci*co <= 4096
  int m = blockIdx.x;
  int tid = threadIdx.x;
  int x = m >> 4, y = m & 15;
  const float* Wsel = (x < 16) ? W1 : W2;
  int xw = x & 15;
  for (int t = tid; t < ci * co; t += 256) {
    int i = t / co, o = t % co;
    size_t wi = ((((size_t)i * co + o) * 16 + xw) * 16 + y) * 2;
    sW[t] = make_float2(Wsel[wi], Wsel[wi + 1]);
  }
  for (int t = tid; t < nb * ci; t += 256) {
    size_t fi = ((size_t)t * NMODE + m) * 2;
    sIn[t] = make_float2(Fin[fi], Fin[fi + 1]);
  }
  __syncthreads();
  for (int t = tid; t < nb * co; t += 256) {
    int bb = t / co, o = t % co;
    float ar = 0.f, ai = 0.f;
    const float2* ib = &sIn[bb * ci];
    for (int i = 0; i < ci; ++i) {
      float2 a = ib[i];
      float2 w = sW[i * co + o];
      ar += a.x * w.x - a.y * w.y;
      ai += a.x * w.y + a.y * w.x;
    }
    size_t o2 = (((size_t)bb * co + o) * NMODE + m) * 2;
    Fout[o2] = ar; Fout[o2 + 1] = ai;
  }
}

// ======================================================================
// 1x1 conv over channels: out[bb,o,h,:] = sum_i in[bb,i,h,:]*w[i,o] + b[o]
// One WG per (bb, h).
// ======================================================================
__global__ __launch_bounds__(256) void k_conv1x1(
    const float* __restrict__ in, const float* __restrict__ w,
    const float* __restrict__ bias, float* __restrict__ out,
    int ci, int co) {
  __shared__ float sRow[4096];   // ci*64 <= 4096
  __shared__ float sW[4096];     // ci*co <= 4096
  __shared__ float sB[64];
  int bb = blockIdx.x >> 6;
  int h  = blockIdx.x & 63;
  int tid = threadIdx.x;
  for (int t = tid; t < ci * 64; t += 256) {
    int i = t >> 6, wc = t & 63;
    sRow[t] = in[((size_t)bb * ci + i) * 4096 + h * 64 + wc];
  }
  for (int t = tid; t < ci * co; t += 256) sW[t] = w[t];
  if (tid < co) sB[tid] = bias[tid];
  __syncthreads();
  for (int t = tid; t < co * 64; t += 256) {
    int o = t >> 6, wc = t & 63;
    float acc = sB[o];
    for (int i = 0; i < ci; ++i) acc += sRow[i * 64 + wc] * sW[i * co + o];
    out[((size_t)bb * co + o) * 4096 + h * 64 + wc] = acc;
  }
}

// ======================================================================
// Truncated inverse rDFT2 + optional (+pre, gelu, +post).  One WG/image.
// ======================================================================
__global__ __launch_bounds__(256) void k_irfft(
    const float* __restrict__ G, const float* __restrict__ pre,
    const float* __restrict__ post, float* __restrict__ out,
    int nimg, int do_gelu) {
  __shared__ float2 sG[512];
  __shared__ float2 sA[1024];
  __shared__ float2 sTw[64];
  int n = blockIdx.x;
  if (n >= nimg) return;
  int tid = threadIdx.x;
  const float* gp = G + (size_t)n * NMODE * 2;
  for (int t = tid; t < 512; t += 256) sG[t] = make_float2(gp[2 * t], gp[2 * t + 1]);
  if (tid < 64) {
    float ang = TWO_PI * (float)tid * (1.0f / 64.0f);
    float s, c; __sincosf(ang, &s, &c);
    sTw[tid] = make_float2(c, s);
  }
  __syncthreads();
  // stage 1: inverse DFT along ky  (A[h][kx] = 1/64 * sum_modes)
  for (int t = tid; t < 1024; t += 256) {
    int h = t >> 4, kx = t & 15;
    float ar = 0.f, ai = 0.f;
    for (int xm = 0; xm < 32; ++xm) {
      int ky = (xm < 16) ? xm : (xm + 32);
      float2 g = sG[xm * 16 + kx];
      float2 tw = sTw[(h * ky) & 63];
      ar += g.x * tw.x - g.y * tw.y;
      ai += g.x * tw.y + g.y * tw.x;
    }
    sA[t] = make_float2(ar * (1.0f / 64.0f), ai * (1.0f / 64.0f));
  }
  __syncthreads();
  // stage 2: inverse real DFT along kx (modes 0..15 only)
  for (int t = tid; t < 4096; t += 256) {
    int h = t >> 6, w = t & 63;
    float v = sA[h * 16].x;
    for (int kx = 1; kx < 16; ++kx) {
      float2 a = sA[h * 16 + kx];
      float2 tw = sTw[(w * kx) & 63];
      v += 2.0f * (a.x * tw.x - a.y * tw.y);
    }
    v *= (1.0f / 64.0f);
    size_t o = (size_t)n * 4096 + t;
    if (pre)  v += pre[o];
    if (do_gelu) v = gelu_f(v);
    if (post) v += post[o];
    out[o] = v;
  }
}

// ======================================================================
// Subdomain amp/phase MLPs + coefficient reduction over subdomains:
// uftc[b,c,x,y] = sum_s F[(b,s,c),x,y] * amp[b,s]*exp(-i(pha0*kx+pha1*ky))
// One WG per batch b.
// ======================================================================
__global__ __launch_bounds__(256) void k_coef(
    const float* __restrict__ F, const float* __restrict__ sd_info,
    const float* __restrict__ aw0, const float* __restrict__ ab0,
    const float* __restrict__ aw1, const float* __restrict__ ab1,
    const float* __restrict__ pw0, const float* __restrict__ pb0,
    const float* __restrict__ pw1, const float* __restrict__ pb1,
    float* __restrict__ uftc) {
  __shared__ float sP0[16], sP1[16], sAm[16];
  int b = blockIdx.x;
  int tid = threadIdx.x;
  if (tid < 16) {
    const float* si = sd_info + ((size_t)b * 16 + tid) * 4;
    float xm0 = si[0], xm1 = si[1], xl0 = si[2], xl1 = si[3];
    float p0 = pb1[0], p1 = pb1[1];
    float am = ab1[0];
    for (int k = 0; k < 32; ++k) {
      float hp = tanhf(xm0 * pw0[k] + xm1 * pw0[32 + k] + pb0[k]);
      p0 += hp * pw1[k * 2 + 0];
      p1 += hp * pw1[k * 2 + 1];
      float ha = tanhf(xl0 * aw0[k] + xl1 * aw0[32 + k] + ab0[k]);
      am += ha * aw1[k];
    }
    sP0[tid] = p0; sP1[tid] = p1; sAm[tid] = am;
  }
  __syncthreads();
  for (int t = tid; t < SDC * NMODE; t += 256) {
    int c = t >> 9;
    int m = t & 511;
    int x = m >> 4, y = m & 15;
    float kx = (float)y;
    float ky = (x < 16) ? (float)x : (float)(x - 32);
    float ar = 0.f, ai = 0.f;
    for (int s = 0; s < 16; ++s) {
      float th = sP0[s] * kx + sP1[s] * ky;
      float sn, cs; __sincosf(th, &sn, &cs);
      float cr = sAm[s] * cs, cim = -sAm[s] * sn;  // amp*exp(-i th)
      size_t fi = ((((size_t)b * 16 + s) * 4 + c) * NMODE + m) * 2;
      float fr = F[fi], fim = F[fi + 1];
      ar += fr * cr - fim * cim;
      ai += fr * cim + fim * cr;
    }
    size_t o = (((size_t)b * 4 + c) * NMODE + m) * 2;
    uftc[o] = ar; uftc[o + 1] = ai;
  }
}

// ======================================================================
// Build WMMA A-operand buffer with Hermitian extension, PAIR-INTERLEAVED
// so each lane's (k, k+1) = (Re, -Im) of one mode is a single b64 load:
//   WB[((b*992 + x*31+y) * 64 + c) * 2 + {0,1}] = {Re(O), -Im(O)}
// ======================================================================
__global__ __launch_bounds__(256) void k_buildW(
    const float* __restrict__ M, float* __restrict__ WB) {
  int idx = blockIdx.x * 256 + threadIdx.x;
  if (idx >= NB * 32 * 31 * 64) return;
  int c = idx & 63;
  int r = idx >> 6;
  int y = r % 31; r /= 31;
  int x = r % 32;
  int b = r / 32;
  float re, im;
  if (y < 16) {
    size_t mi = (((size_t)b * 64 + c) * NMODE + x * 16 + y) * 2;
    re = M[mi]; im = M[mi + 1];
  } else {
    int j = y - 16;
    size_t mi = (((size_t)b * 64 + c) * NMODE + (31 - x) * 16 + (15 - j)) * 2;
    re = M[mi]; im = -M[mi + 1];       // conj
  }
  size_t wo = (((size_t)b * NPAIR + (size_t)x * 31 + y) * 64 + c) * 2;
  WB[wo]     = re;
  WB[wo + 1] = -im;
}

// ======================================================================
// WMMA non-uniform spectral evaluation (dominant ~33 GFLOP):
// ubcn[b,c,n] = Re( sum_{x,y} O[b,c,x,y] * exp(2pi i (px*k1[x]+py*k2[y])) )
//               + u3[b,c,n]
// WG = (b, 64-point tile); 8 waves; each wave owns (Mblk, 2 N-blocks).
// A: b64 loads from L2 (+prefetch of next chunk).
// B: cos/sin basis in LDS, PAIR-INTERLEAVED [pair][n][{cos,sin}] so each
//    lane's B operand is one aligned ds_load_b64 (no repack movs).
// Output channel-pair interleaved: ubcn[((b*32 + c/2)*NPTS + n)*2 + (c&1)]
// ======================================================================
__global__ __launch_bounds__(256) void k_eval(
    const float* __restrict__ WB, const float* __restrict__ xy,
    const float* __restrict__ b4w, const float* __restrict__ b4b,
    float* __restrict__ ubcn) {
  __shared__ float sS[62 * 64 * 2];   // basis chunk: [pair][n][{cos,sin}]
  __shared__ float sPx[64], sPy[64];
  int b    = blockIdx.x >> 8;
  int tile = (blockIdx.x & 255) * 64;
  int tid  = threadIdx.x;
  int lane = tid & 31;
  int wid  = tid >> 5;
  int Mblk = wid >> 1;
  int Npair = wid & 1;
  int lm = lane & 15;
  int khalf = lane >> 4;               // 0 or 1 -> pair offset within k-step

  if (tid < 64) {
    size_t pi = ((size_t)b * NPTS + tile + tid) * 2;
    sPx[tid] = xy[pi];
    sPy[tid] = xy[pi + 1];
  }

  v8f acc0 = (v8f)0.0f;
  v8f acc1 = (v8f)0.0f;

  const float* Wbase = WB + (size_t)b * NPAIR * 128;   // 64 ch * 2 floats/pair

  for (int cc = 0; cc < 16; ++cc) {      // x-chunks of 2 (32 x values)
    __syncthreads();
    // prefetch next chunk's A-operand stream (31 KB) into cache while the
    // sincos basis build runs (global_prefetch_b8; one cacheline per lane)
    if (cc < 15) {
      const char* nxt = (const char*)(Wbase + (size_t)(cc + 1) * 62 * 128);
      __builtin_prefetch(nxt + (size_t)tid * 128, 0, 1);
    }
    // build basis chunk: 62 (x,y) pairs * 64 points
    for (int p = tid; p < 62 * 64; p += 256) {
      int n = p & 63;
      int q = p >> 6;                    // 0..61
      int xl = q / 31, yy = q % 31;
      int x = cc * 2 + xl;
      float k1 = (x < 16) ? (float)x : (float)(x - 32);
      float k2 = (yy < 16) ? (float)yy : (float)(yy - 31);
      float th = TWO_PI * (sPx[n] * k1 + sPy[n] * k2);
      float sn, cs; __sincosf(th, &sn, &cs);
      sS[(q * 64 + n) * 2 + 0] = cs;
      sS[(q * 64 + n) * 2 + 1] = sn;
    }
    __syncthreads();
    const float* Wc = Wbase + (size_t)cc * 62 * 128;   // chunk pair base
    int cch = Mblk * 16 + lm;
    int nA = Npair * 32 + lm;            // point column for N-block pair base
    for (int ks = 0; ks < 31; ++ks) {    // 124 / 4
      int gp = ks * 2 + khalf;           // basis pair index for this lane
      v2f a = *(const v2f*)(Wc + ((size_t)gp * 64 + cch) * 2);       // b64
      v2f b0 = *(const v2f*)(sS + ((size_t)gp * 64 + nA) * 2);       // b64
      v2f b1 = *(const v2f*)(sS + ((size_t)gp * 64 + nA + 16) * 2);  // b64
      acc0 = wmma4(a, b0, acc0);
      acc1 = wmma4(a, b1, acc1);
    }
  }
  // store D tiles + fused u3 = xy @ b4_w + b4_b  (channel-pair interleaved)
  int rowoff = (lane >> 4) << 3;
  for (int which = 0; which < 2; ++which) {
    v8f* accp = which ? &acc1 : &acc0;
    int nblk = Npair * 2 + which;
    int nl = nblk * 16 + lm;
    int n = tile + nl;
    float px = sPx[nl], py = sPy[nl];
    for (int r = 0; r < 8; ++r) {
      int c = Mblk * 16 + r + rowoff;
      float u3 = px * b4w[c] + py * b4w[64 + c] + b4b[c];
      size_t oo = (((size_t)b * 32 + (c >> 1)) * NPTS + n) * 2 + (c & 1);
      ubcn[oo] = (*accp)[r] + u3;
    }
  }
}

// ======================================================================
// WMMA head: h = gelu(v @ fc1 + b1); out = h @ fc2 + b2, fc2 fused as
// per-lane weighted sum + wave32 shuffle reduction. One wave / 16 points.
// A operand: b64 (channel-pair interleaved ubcn).
// B operand: fc1 weights staged pair-interleaved in LDS -> ds_load_b64.
// ======================================================================
__global__ __launch_bounds__(256) void k_head(
    const float* __restrict__ ubcn, const float* __restrict__ fc1w,
    const float* __restrict__ fc1b, const float* __restrict__ fc2w,
    const float* __restrict__ fc2b, float* __restrict__ out) {
  __shared__ float sW[64 * 128];   // [cpair][out][{c_even, c_odd}]
  __shared__ float sB1[128];
  __shared__ float sW2[128];
  int tid = threadIdx.x;
  int lane = tid & 31;
  int wid = tid >> 5;
  int lm = lane & 15;
  int khalf = lane >> 4;           // 0 or 1
  for (int t = tid; t < 64 * 128; t += 256) {
    int c = t >> 7, o = t & 127;
    sW[(((c >> 1) * 128) + o) * 2 + (c & 1)] = fc1w[t];
  }
  if (tid < 128) { sB1[tid] = fc1b[tid]; sW2[tid] = fc2w[tid]; }
  __syncthreads();

  size_t gp = (size_t)blockIdx.x * 128 + (size_t)wid * 16;  // global point base
  int b = (int)(gp >> 14);
  int n0 = (int)(gp & 16383);

  float psum[8];
  for (int r = 0; r < 8; ++r) psum[r] = 0.f;

  for (int j = 0; j < 8; ++j) {          // 128 outputs in 8 N-blocks
    v8f d = (v8f)0.0f;
    int ncol = j * 16 + lm;
    for (int k = 0; k < 16; ++k) {       // K = 64 channels
      int cp = k * 2 + khalf;            // channel pair index
      size_t base = (((size_t)b * 32 + cp) * NPTS + n0 + lm) * 2;
      v2f a = *(const v2f*)(ubcn + base);                 // b64
      v2f bb = *(const v2f*)(sW + ((size_t)cp * 128 + ncol) * 2);  // b64
      d = wmma4(a, bb, d);
    }
    float bias = sB1[ncol];
    float w2 = sW2[ncol];
    for (int r = 0; r < 8; ++r) {
      float h = gelu_f(d[r] + bias);
      psum[r] += h * w2;
    }
  }
  // reduce over the 16 lanes of each half (N dimension of fc1 output)
  for (int off = 1; off < 16; off <<= 1)
    for (int r = 0; r < 8; ++r)
      psum[r] += __shfl_xor(psum[r], off, 32);
  if (lm == 0) {
    float b2 = fc2b[0];
    int rowoff = (lane >> 4) << 3;
    for (int r = 0; r < 8; ++r)
      out[gp + r + rowoff] = psum[r] + b2;
  }
}

// ======================================================================
// Host-side orchestration
// ======================================================================
extern "C" void kernel_launch(void* const* d_in, const int* in_sizes, int n_in,
                              void* d_out, int out_size, void* d_ws, size_t ws_size,
                              hipStream_t stream) {
  (void)in_sizes; (void)n_in; (void)out_size; (void)ws_size;
  const float* u_g     = (const float*)d_in[0];
  const float* u_sd    = (const float*)d_in[1];
  const float* sd_info = (const float*)d_in[2];
  const float* xy      = (const float*)d_in[3];
  const float* enc_fc_w = (const float*)d_in[4];
  const float* enc_fc_b = (const float*)d_in[5];
  const float* enc_cw1  = (const float*)d_in[6];
  const float* enc_cw2  = (const float*)d_in[7];
  const float* enc_w0_w = (const float*)d_in[8];
  const float* enc_w0_b = (const float*)d_in[9];
  const float* amp_w0 = (const float*)d_in[10];
  const float* amp_b0 = (const float*)d_in[11];
  const float* amp_w1 = (const float*)d_in[12];
  const float* amp_b1 = (const float*)d_in[13];
  const float* pha_w0 = (const float*)d_in[14];
  const float* pha_b0 = (const float*)d_in[15];
  const float* pha_w1 = (const float*)d_in[16];
  const float* pha_b1 = (const float*)d_in[17];
  const float* dec_cw1 = (const float*)d_in[18];
  const float* dec_cw2 = (const float*)d_in[19];
  const float* cw1[5] = { (const float*)d_in[20], (const float*)d_in[22],
                          (const float*)d_in[24], (const float*)d_in[26],
                          (const float*)d_in[28] };
  const float* cw2[5] = { (const float*)d_in[21], (const float*)d_in[23],
                          (const float*)d_in[25], (const float*)d_in[27],
                          (const float*)d_in[29] };
  const float* wlw[4] = { (const float*)d_in[30], (const float*)d_in[32],
                          (const float*)d_in[34], (const float*)d_in[36] };
  const float* wlb[4] = { (const float*)d_in[31], (const float*)d_in[33],
                          (const float*)d_in[35], (const float*)d_in[37] };
  const float* b4_w  = (const float*)d_in[38];
  const float* b4_b  = (const float*)d_in[39];
  const float* fc0_w = (const float*)d_in[40];
  const float* fc0_b = (const float*)d_in[41];
  const float* fc1_w = (const float*)d_in[42];
  const float* fc1_b = (const float*)d_in[43];
  const float* fc2_w = (const float*)d_in[44];
  const float* fc2_b = (const float*)d_in[45];
  float* outp = (float*)d_out;

  char* ws = (char*)d_ws;
  float* A    = (float*)(ws + ((size_t)0  << 20));   // spatial (<= 512 imgs)
  float* Bt   = (float*)(ws + ((size_t)16 << 20));   // spatial tmp
  float* M0   = (float*)(ws + ((size_t)32 << 20));   // modes
  float* M1   = (float*)(ws + ((size_t)48 << 20));   // mixed modes
  float* USDG = (float*)(ws + ((size_t)64 << 20));   // decoded subdomain field
  float* UFTC = (float*)(ws + ((size_t)80 << 20));   // reduced coefficients
  float* WB   = (float*)(ws + ((size_t)96 << 20));   // WMMA A operand
  float* UB   = (float*)(ws + ((size_t)112 << 20));  // ubcn (b, c/2, n, 2)

  const int NIMG = 512;   // 128*4 (encoder) == 8*64 (main) spatial images

  // ---- encoder ----
  k_embed_sd<<<8192, 256, 0, stream>>>(u_sd, sd_info, enc_fc_w, enc_fc_b, A);
  k_rfft<<<NIMG, 256, 0, stream>>>(A, M0, NIMG);
  k_specmix<<<NMODE, 256, 0, stream>>>(M0, enc_cw1, enc_cw2, M1, 128, 4, 4);
  k_conv1x1<<<128 * 64, 256, 0, stream>>>(A, enc_w0_w, enc_w0_b, Bt, 4, 4);
  k_irfft<<<NIMG, 256, 0, stream>>>(M1, Bt, nullptr, A, NIMG, 1);
  k_rfft<<<NIMG, 256, 0, stream>>>(A, M0, NIMG);   // uft of encoder output
  k_coef<<<NB, 256, 0, stream>>>(M0, sd_info, amp_w0, amp_b0, amp_w1, amp_b1,
                                 pha_w0, pha_b0, pha_w1, pha_b1, UFTC);
  k_specmix<<<NMODE, 256, 0, stream>>>(UFTC, dec_cw1, dec_cw2, M1, 8, 4, 64);
  k_irfft<<<NIMG, 256, 0, stream>>>(M1, nullptr, nullptr, USDG, NIMG, 0);

  // ---- main branch ----
  k_embed_g<<<8192, 256, 0, stream>>>(u_g, fc0_w, fc0_b, A);
  for (int L = 0; L < 4; ++L) {
    k_rfft<<<NIMG, 256, 0, stream>>>(A, M0, NIMG);
    k_specmix<<<NMODE, 256, 0, stream>>>(M0, cw1[L], cw2[L], M1, 8, 64, 64);
    k_conv1x1<<<8 * 64, 256, 0, stream>>>(A, wlw[L], wlb[L], Bt, 64, 64);
    const float* post = (L == 0) ? USDG : nullptr;
    k_irfft<<<NIMG, 256, 0, stream>>>(M1, Bt, post, A, NIMG, 1);
  }
  // layer 4: spectral only -> Hermitian-extended W operand (pair-interleaved)
  k_rfft<<<NIMG, 256, 0, stream>>>(A, M0, NIMG);
  k_specmix<<<NMODE, 256, 0, stream>>>(M0, cw1[4], cw2[4], M1, 8, 64, 64);
  k_buildW<<<(NB * 32 * 31 * 64 + 255) / 256, 256, 0, stream>>>(M1, WB);

  // ---- WMMA non-uniform evaluation + head ----
  k_eval<<<NB * (NPTS / 64), 256, 0, stream>>>(WB, xy, b4_w, b4_b, UB);
  k_head<<<(NB * NPTS) / 128, 256, 0, stream>>>(UB, fc1_w, fc1_b, fc2_w, fc2_b, outp);
}